// ViewLearner_89292370084347
// MI455X (gfx1250) — compile-verified
//
#include <hip/hip_runtime.h>
#include <hip/hip_bf16.h>

// Sizes from the reference: N=50000, D=128, E=800000, H=64.
#define FEAT_D   128
#define HID_H    64
#define BIAS_K   1e-4f
#define MT       3          // row-tiles (16 rows each) per wave in kernel 1
#define W1T_STRIDE 260      // padded row stride of transposed W1 in LDS
// ws layout: P[N*64] fp32, then Q[N*64] fp32  (2*50000*64*4 = 25.6 MB)

typedef float v2f __attribute__((ext_vector_type(2)));
typedef float v8f __attribute__((ext_vector_type(8)));

// ---------------------------------------------------------------------------
// Kernel 1: P = node_emb @ W1[:128,:], Q = node_emb @ W1[128:,:]
// One wave computes MT=3 16x64 tiles of BOTH P and Q with
// V_WMMA_F32_16X16X4_F32, so each B fragment (shared by all tiles) feeds
// 24 WMMAs per 8 LDS b64 loads. W1 is staged TRANSPOSED in LDS
// ([col][row], stride 260) so a B frag is one aligned ds_load_b64, and the
// word-bank pattern (4*ncol + krow) mod 64 covers all 64 banks conflict-free.
// Out-of-range tiles are clamped (duplicate idempotent writes) so EXEC stays
// all-ones for WMMA.
// ---------------------------------------------------------------------------
__global__ __launch_bounds__(256) void
vl_mlp1_gemm(const float* __restrict__ node_emb,
             const float* __restrict__ W1,
             float* __restrict__ P,
             float* __restrict__ Q,
             int nrows)
{
    __shared__ float w1t[HID_H * W1T_STRIDE];   // ~65 KB of 320 KB WGP LDS

    const int tid = threadIdx.x;
    // cooperative load + transpose: 4096 float4s of W1[256][64]
    for (int j = tid; j < (256 * HID_H) / 4; j += 256) {
        const float4 v = ((const float4*)W1)[j];
        const int r  = j >> 4;          // W1 row 0..255
        const int c0 = (j & 15) * 4;    // W1 col 0..60
        w1t[(c0 + 0) * W1T_STRIDE + r] = v.x;
        w1t[(c0 + 1) * W1T_STRIDE + r] = v.y;
        w1t[(c0 + 2) * W1T_STRIDE + r] = v.z;
        w1t[(c0 + 3) * W1T_STRIDE + r] = v.w;
    }
    __syncthreads();

    const int wave  = tid >> 5;
    const int lane  = tid & 31;
    const int m     = lane & 15;        // A row this lane supplies
    const int khalf = lane >> 4;        // 0: K={0,1}, 1: K={2,3}
    const int ncol  = lane & 15;        // B/D column this lane supplies

    const int ntiles   = nrows >> 4;    // 3125 (nrows % 16 == 0)
    const int tilebase = (blockIdx.x * 8 + wave) * MT;
    if (tilebase >= ntiles) return;     // wave-uniform

    int rowstart[MT];
    const float* arow[MT];
    #pragma unroll
    for (int mt = 0; mt < MT; ++mt) {
        int t = tilebase + mt;
        if (t >= ntiles) t = ntiles - 1;            // clamp: idempotent dup
        rowstart[mt] = t * 16;
        arow[mt] = node_emb + (size_t)(t * 16 + m) * FEAT_D;
    }

    v8f acc[MT][8];                     // [mt][h*4+nt]: h=0 -> P, h=1 -> Q
    #pragma unroll
    for (int mt = 0; mt < MT; ++mt)
        #pragma unroll
        for (int i = 0; i < 8; ++i) acc[mt][i] = (v8f){0,0,0,0,0,0,0,0};

    for (int k = 0; k < FEAT_D; k += 4) {
        // A frags: 16x4 f32; VGPR0=K(khalf*2), VGPR1=K(khalf*2+1)
        v2f a[MT];
        #pragma unroll
        for (int mt = 0; mt < MT; ++mt)
            a[mt] = *(const v2f*)&arow[mt][k + khalf * 2];

        // B frags: one aligned b64 from transposed LDS per (h, nt)
        v2f b[8];
        #pragma unroll
        for (int h = 0; h < 2; ++h) {
            const int krow = h * FEAT_D + k + khalf * 2;
            #pragma unroll
            for (int nt = 0; nt < 4; ++nt)
                b[h * 4 + nt] =
                    *(const v2f*)&w1t[(nt * 16 + ncol) * W1T_STRIDE + krow];
        }

        #pragma unroll
        for (int mt = 0; mt < MT; ++mt)
            #pragma unroll
            for (int f = 0; f < 8; ++f)
                acc[mt][f] = __builtin_amdgcn_wmma_f32_16x16x4_f32(
                    false, a[mt], false, b[f],
                    (short)0, acc[mt][f], false, false);
    }

    // D layout 16x16 f32: VGPR v -> row (v + khalf*8), col = ncol
    #pragma unroll
    for (int mt = 0; mt < MT; ++mt) {
        #pragma unroll
        for (int h = 0; h < 2; ++h) {
            float* out = (h == 0 ? P : Q) + (size_t)rowstart[mt] * HID_H;
            #pragma unroll
            for (int nt = 0; nt < 4; ++nt) {
                v8f c = acc[mt][h * 4 + nt];
                #pragma unroll
                for (int v = 0; v < 8; ++v)
                    out[(size_t)(v + khalf * 8) * HID_H + nt * 16 + ncol] = c[v];
            }
        }
    }
}

// ---------------------------------------------------------------------------
// Kernel 2: per-edge  h = relu(P[src]+Q[dst]+b1);  logit = h.W2 + b2;
//           eps-gate + sigmoid.  4 lanes cooperate per edge; chunk index
//           (i*4 + sub) makes the 4 lanes read one contiguous 64B line of
//           the L2-resident P/Q rows per load.  Wave32 shfl_xor reduction.
// ---------------------------------------------------------------------------
__global__ __launch_bounds__(256) void
vl_edge_gate(const float* __restrict__ P,
             const float* __restrict__ Q,
             const float* __restrict__ b1,
             const float* __restrict__ W2,
             const float* __restrict__ b2,
             const int* __restrict__ edge_index,   // [2,E] flat
             const float* __restrict__ eps_u,      // [E]
             float* __restrict__ out,              // [E]
             int nedges)
{
    __shared__ float w2s[HID_H];
    __shared__ float b1s[HID_H];
    if (threadIdx.x < HID_H) {
        w2s[threadIdx.x] = W2[threadIdx.x];
        b1s[threadIdx.x] = b1[threadIdx.x];
    }
    __syncthreads();

    const long long gid = (long long)blockIdx.x * blockDim.x + threadIdx.x;
    const int edge = (int)(gid >> 2);
    const int sub  = (int)(gid & 3);
    if (edge >= nedges) return;

    const int s = edge_index[edge];
    const int d = edge_index[nedges + edge];

    const float4* p4 = (const float4*)(P + (size_t)s * HID_H);
    const float4* q4 = (const float4*)(Q + (size_t)d * HID_H);

    float partial = 0.0f;
    #pragma unroll
    for (int i = 0; i < 4; ++i) {
        const int ch = i * 4 + sub;            // float4 chunk 0..15
        const float4 pv = p4[ch];
        const float4 qv = q4[ch];
        const float4 wv = ((const float4*)w2s)[ch];
        const float4 bv = ((const float4*)b1s)[ch];
        partial = fmaf(fmaxf(pv.x + qv.x + bv.x, 0.0f), wv.x, partial);
        partial = fmaf(fmaxf(pv.y + qv.y + bv.y, 0.0f), wv.y, partial);
        partial = fmaf(fmaxf(pv.z + qv.z + bv.z, 0.0f), wv.z, partial);
        partial = fmaf(fmaxf(pv.w + qv.w + bv.w, 0.0f), wv.w, partial);
    }
    // reduce the 4 sub-lane partials (lanes 4k..4k+3 of the wave32)
    partial += __shfl_xor(partial, 1, 32);
    partial += __shfl_xor(partial, 2, 32);

    if (sub == 0) {
        const float logit = partial + b2[0];
        const float u     = eps_u[edge];
        // eps = (BIAS - (1-BIAS))*u + (1-BIAS)  in (BIAS, 1-BIAS)
        const float eps   = (2.0f * BIAS_K - 1.0f) * u + (1.0f - BIAS_K);
        const float gate  = (__logf(eps) - __logf(1.0f - eps) + logit) * 0.5f;
        out[edge] = 1.0f / (1.0f + __expf(-gate));   // sigmoid; where() is a no-op
    }
}

// ---------------------------------------------------------------------------
extern "C" void kernel_launch(void* const* d_in, const int* in_sizes, int n_in,
                              void* d_out, int out_size, void* d_ws, size_t ws_size,
                              hipStream_t stream)
{
    const float* node_emb = (const float*)d_in[0];   // [N,128]
    const float* W1       = (const float*)d_in[1];   // [256,64]
    const float* b1       = (const float*)d_in[2];   // [64]
    const float* W2       = (const float*)d_in[3];   // [64,1]
    const float* b2       = (const float*)d_in[4];   // [1]
    const int*   eidx     = (const int*)d_in[5];     // [2,E]
    const float* eps_u    = (const float*)d_in[6];   // [E,1]
    float*       out      = (float*)d_out;           // [E]

    const int Nn = in_sizes[0] / FEAT_D;             // 50000
    const int Ee = in_sizes[6];                      // 800000

    float* Pbuf = (float*)d_ws;
    float* Qbuf = Pbuf + (size_t)Nn * HID_H;

    // Kernel 1: 3125 row-tiles, 8 waves/block, MT tiles per wave
    const int ntiles  = (Nn + 15) / 16;
    const int tiles_per_block = 8 * MT;
    const int blocks1 = (ntiles + tiles_per_block - 1) / tiles_per_block;
    hipLaunchKernelGGL(vl_mlp1_gemm, dim3(blocks1), dim3(256), 0, stream,
                       node_emb, W1, Pbuf, Qbuf, Nn);

    // Kernel 2: 4 lanes per edge
    const long long threads2 = (long long)Ee * 4;
    const int blocks2 = (int)((threads2 + 255) / 256);
    hipLaunchKernelGGL(vl_edge_gate, dim3(blocks2), dim3(256), 0, stream,
                       Pbuf, Qbuf, b1, W2, b2, eidx, eps_u, out, Ee);
}